// GCN_5153960755743
// MI455X (gfx1250) — compile-verified
//
#include <hip/hip_runtime.h>
#include <hip/hip_bf16.h>

typedef __attribute__((ext_vector_type(2))) float v2f;
typedef __attribute__((ext_vector_type(8))) float v8f;
typedef __attribute__((ext_vector_type(4))) int   v4i;
typedef __attribute__((address_space(1))) v4i*    gv4i_p;  // global int4*
typedef __attribute__((address_space(3))) v4i*    lv4i_p;  // LDS int4*

#define LDS_STRIDE 144   // 128 + 16 pad: keeps the two half-wave row reads on disjoint bank groups

#if defined(__has_builtin)
#if __has_builtin(__builtin_amdgcn_global_load_async_to_lds_b128)
#define HAVE_ASYNC_LDS 1
#endif
#endif

// ---------------------------------------------------------------------------
// zero a float region
// ---------------------------------------------------------------------------
__global__ void gcn_zero_kernel(float* __restrict__ p, long n) {
    long i = (long)blockIdx.x * blockDim.x + threadIdx.x;
    if (i < n) p[i] = 0.0f;
}

// ---------------------------------------------------------------------------
// degree accumulation: deg_out[src[e]] += 1, deg_in[dst[e]] += 1
// ---------------------------------------------------------------------------
__global__ void gcn_degree_kernel(const int* __restrict__ src,
                                  const int* __restrict__ dst,
                                  float* __restrict__ deg_out,
                                  float* __restrict__ deg_in, int E) {
    int i = blockIdx.x * blockDim.x + threadIdx.x;
    if (i < E) {
        atomicAdd(&deg_out[src[i]], 1.0f);
        atomicAdd(&deg_in[dst[i]], 1.0f);
    }
}

// ---------------------------------------------------------------------------
// norm = rsqrt(max(deg, 1))
// ---------------------------------------------------------------------------
__global__ void gcn_norm_kernel(const float* __restrict__ deg,
                                float* __restrict__ norm, int N) {
    int i = blockIdx.x * blockDim.x + threadIdx.x;
    if (i < N) norm[i] = rsqrtf(fmaxf(deg[i], 1.0f));
}

// ---------------------------------------------------------------------------
// H = (X * norm_row) @ W   via V_WMMA_F32_16X16X4_F32
//   X: [N,128], W: [128,128] row-major (k -> n), H: [N,128]
//   block = 128 threads = 4 waves; each wave computes a 16x128 slab
//   W staged into LDS via GLOBAL_LOAD_ASYNC_TO_LDS_B128 (L2 -> LDS DMA)
// ---------------------------------------------------------------------------
__global__ __launch_bounds__(128) void gcn_gemm_wmma_kernel(
    const float* __restrict__ X, const float* __restrict__ norm,
    const float* __restrict__ W, float* __restrict__ H, int N) {
    __shared__ float sW[128 * LDS_STRIDE];

    const int tid  = threadIdx.x;
    const int wave = tid >> 5;
    const int lane = tid & 31;

#ifdef HAVE_ASYNC_LDS
    // async DMA of W (4096 x b128) directly into padded LDS, no VGPR round-trip
    for (int i = tid; i < 4096; i += 128) {
        const int r  = i >> 5;          // W row (32 float4 per row)
        const int c4 = (i & 31) << 2;   // column of this float4
        __builtin_amdgcn_global_load_async_to_lds_b128(
            (gv4i_p)(void*)(W + r * 128 + c4),
            (lv4i_p)(void*)(sW + r * LDS_STRIDE + c4),
            0, 0);
    }
#if __has_builtin(__builtin_amdgcn_s_wait_asynccnt)
    __builtin_amdgcn_s_wait_asynccnt(0);
#else
    asm volatile("s_wait_asynccnt 0x0" ::: "memory");
#endif
#else
    for (int i = tid; i < 128 * 128; i += 128) {
        int r = i >> 7, c = i & 127;
        sW[r * LDS_STRIDE + c] = W[i];
    }
#endif
    __syncthreads();

    const int   row0  = (blockIdx.x * 4 + wave) * 16;
    const int   m     = lane & 15;           // matrix row / col within tile
    const int   row   = row0 + m;
    const bool  valid = row < N;
    const float scale = valid ? norm[row] : 0.0f;   // scale==0 zeroes OOB A rows
    const int   khalf = (lane >> 4) * 2;     // lanes 16-31 hold K+2/K+3

    const float* xrow = X + (size_t)(valid ? row : 0) * 128;

    v8f acc[8] = {};   // 8 N-tiles of 16 columns each

    for (int ks = 0; ks < 32; ++ks) {
        const int kbase = ks * 4;
        // A fragment: VGPR j, lane L -> A[L%16][kbase + j + 2*(L>=16)]
        v2f a = *(const v2f*)(xrow + kbase + khalf);
        a.x *= scale;
        a.y *= scale;
#pragma unroll
        for (int t = 0; t < 8; ++t) {
            const int n0 = t * 16;
            // B fragment: VGPR j, lane L -> W[kbase + j + 2*(L>=16)][n0 + L%16]
            v2f b;
            b.x = sW[(kbase + khalf + 0) * LDS_STRIDE + n0 + m];
            b.y = sW[(kbase + khalf + 1) * LDS_STRIDE + n0 + m];
            acc[t] = __builtin_amdgcn_wmma_f32_16x16x4_f32(
                false, a, false, b, (short)0, acc[t], false, false);
        }
    }

    // D layout: VGPR v, lanes 0-15 -> M=v, lanes 16-31 -> M=v+8; N = lane%16
    const int nCol = lane & 15;
    const int mOff = (lane >> 4) * 8;
#pragma unroll
    for (int v = 0; v < 8; ++v) {
        const int r = row0 + mOff + v;
        if (r < N) {
            float* hr = H + (size_t)r * 128 + nCol;
#pragma unroll
            for (int t = 0; t < 8; ++t) hr[t * 16] = acc[t][v];
        }
    }
}

// ---------------------------------------------------------------------------
// SpMM scatter: agg[dst[e]] += h[src[e]]  (one wave per edge, float4/lane)
// h and agg are each 25.6 MB -> both L2-resident on MI455X (192 MB L2)
// ---------------------------------------------------------------------------
__global__ __launch_bounds__(256) void gcn_scatter_kernel(
    const float* __restrict__ H, const int* __restrict__ src,
    const int* __restrict__ dst, float* __restrict__ agg, int E) {
    const int edge = (int)((blockIdx.x * 256u + threadIdx.x) >> 5);
    if (edge >= E) return;
    const int lane = threadIdx.x & 31;
    const int s = src[edge];
    const int d = dst[edge];
    const float4 v = *(const float4*)(H + (size_t)s * 128 + lane * 4);
    float* out = agg + (size_t)d * 128 + lane * 4;
    atomicAdd(out + 0, v.x);
    atomicAdd(out + 1, v.y);
    atomicAdd(out + 2, v.z);
    atomicAdd(out + 3, v.w);
}

// ---------------------------------------------------------------------------
// out = relu(agg * norm_in[row] + b[col])   (float4 per thread)
// ---------------------------------------------------------------------------
__global__ void gcn_finalize_kernel(const float* __restrict__ agg,
                                    const float* __restrict__ norm_in,
                                    const float* __restrict__ b,
                                    float* __restrict__ out, int N) {
    const long i = (long)blockIdx.x * blockDim.x + threadIdx.x;  // float4 index
    const long total = (long)N * 32;
    if (i >= total) return;
    const int r  = (int)(i >> 5);
    const int c4 = (int)(i & 31) * 4;
    const float s = norm_in[r];
    const float4 a  = *(const float4*)(agg + (size_t)r * 128 + c4);
    const float4 bb = *(const float4*)(b + c4);
    float4 o;
    o.x = fmaxf(a.x * s + bb.x, 0.0f);
    o.y = fmaxf(a.y * s + bb.y, 0.0f);
    o.z = fmaxf(a.z * s + bb.z, 0.0f);
    o.w = fmaxf(a.w * s + bb.w, 0.0f);
    *(float4*)(out + (size_t)r * 128 + c4) = o;
}

// ---------------------------------------------------------------------------
extern "C" void kernel_launch(void* const* d_in, const int* in_sizes, int n_in,
                              void* d_out, int out_size, void* d_ws, size_t ws_size,
                              hipStream_t stream) {
    const float* x0  = (const float*)d_in[0];
    const int*   src = (const int*)d_in[1];
    const int*   dst = (const int*)d_in[2];
    const float* W1  = (const float*)d_in[3];
    const float* b1  = (const float*)d_in[4];
    const float* W2  = (const float*)d_in[5];
    const float* b2  = (const float*)d_in[6];
    float* out = (float*)d_out;

    const int N = in_sizes[0] / 128;
    const int E = in_sizes[1];

    // workspace layout (floats)
    float* ws       = (float*)d_ws;
    float* deg_out  = ws;
    float* deg_in   = ws + (size_t)N;
    float* norm_out = ws + (size_t)2 * N;
    float* norm_in  = ws + (size_t)3 * N;
    float* h        = ws + (size_t)4 * N;
    float* agg      = h + (size_t)N * 128;
    float* x1       = agg + (size_t)N * 128;

    const long feat = (long)N * 128;

    // --- degrees & norms (shared by both layers) ---
    {
        long n = 2L * N;
        gcn_zero_kernel<<<(int)((n + 255) / 256), 256, 0, stream>>>(deg_out, n);
    }
    gcn_degree_kernel<<<(E + 255) / 256, 256, 0, stream>>>(src, dst, deg_out, deg_in, E);
    gcn_norm_kernel<<<(2 * N + 255) / 256, 256, 0, stream>>>(deg_out, norm_out, 2 * N);
    // (deg_out/deg_in contiguous -> one launch computes norm_out and norm_in)

    const int gemmBlocks = (N + 63) / 64;  // 4 waves x 16 rows per block

    // --- layer 1 ---
    gcn_gemm_wmma_kernel<<<gemmBlocks, 128, 0, stream>>>(x0, norm_out, W1, h, N);
    gcn_zero_kernel<<<(int)((feat + 255) / 256), 256, 0, stream>>>(agg, feat);
    gcn_scatter_kernel<<<(E + 7) / 8, 256, 0, stream>>>(h, src, dst, agg, E);
    gcn_finalize_kernel<<<(int)((feat / 4 + 255) / 256), 256, 0, stream>>>(agg, norm_in, b1, x1, N);

    // --- layer 2 ---
    gcn_gemm_wmma_kernel<<<gemmBlocks, 128, 0, stream>>>(x1, norm_out, W2, h, N);
    gcn_zero_kernel<<<(int)((feat + 255) / 256), 256, 0, stream>>>(agg, feat);
    gcn_scatter_kernel<<<(E + 7) / 8, 256, 0, stream>>>(h, src, dst, agg, E);
    gcn_finalize_kernel<<<(int)((feat / 4 + 255) / 256), 256, 0, stream>>>(agg, norm_in, b2, out, N);
}